// Model_11570641895529
// MI455X (gfx1250) — compile-verified
//
#include <hip/hip_runtime.h>
#include <cfloat>
#include <cmath>
#include <cstdint>

// Problem constants (fixed by the reference).
constexpr int   Bb   = 32;
constexpr int   Tt   = 512;
constexpr int   Cc   = 4233;
constexpr int   Uu   = 64;
constexpr int   LABT = 2 * Uu + 1;   // 129
constexpr int   BPS  = 132;          // padded byte stride for packed-bp rows
constexpr float PADV = -FLT_MAX;     // jnp.finfo(float32).min

// ---------------------------------------------------------------------------
// Per-lane async copy: 4 bytes from the global address gsrc to the LDS address ldst, tracked by ASYNCcnt.
// AS(3) pointer value is the LDS-base-relative byte offset (same as ds_* addr).
// ---------------------------------------------------------------------------
__device__ __forceinline__ void async_copy_b32(float* ldst, const float* gsrc)
{
    unsigned lds_off = (unsigned)(uintptr_t)(__attribute__((address_space(3))) float*)ldst;
    asm volatile("global_load_async_to_lds_b32 %0, %1, off"
                 :: "v"(lds_off), "v"(gsrc)
                 : "memory");
}

// ---------------------------------------------------------------------------
// Kernel 1: fully parallel gather of emission log-probs.
// emit[b][t][j] = logit[b][t][ext_idx[b][j]]
// ---------------------------------------------------------------------------
__global__ void __launch_bounds__(256) gather_emit_kernel(
    const float* __restrict__ logit,   // [B,T,C]
    const int*   __restrict__ label,   // [U,B]
    const int*   __restrict__ blankp,  // scalar on device
    float*       __restrict__ emit)    // [B,T,LABT]
{
    const int total = Bb * Tt * LABT;
    int i = blockIdx.x * blockDim.x + threadIdx.x;
    if (i >= total) return;
    const int j  = i % LABT;
    const int bt = i / LABT;
    const int t  = bt % Tt;
    const int b  = bt / Tt;
    const int blank = *blankp;

    int extv = blank;
    if (j & 1) {
        const int lv = label[(j >> 1) * Bb + b];
        extv = (lv == blank) ? -1 : lv;
    }
    int idx = extv % Cc;          // mimic Python modulo for negative pad label
    if (idx < 0) idx += Cc;

    emit[i] = logit[((size_t)b * Tt + t) * (size_t)Cc + idx];
}

// ---------------------------------------------------------------------------
// Kernel 2: fused DP + loss + Viterbi backtrack + one-hot alignment writeout.
// One block per batch element; 256 threads (8 wave32). labT dim on threads.
// PRE: emit rows staged via GLOBAL_LOAD_ASYNC_TO_LDS_B32 into a 4-deep ring
//      (depth-3 latency hiding via ASYNCcnt in-order completion).
// ---------------------------------------------------------------------------
template <bool PRE>
__global__ void __launch_bounds__(256) ctc_fa_kernel(
    const float* __restrict__ logit,   // [B,T,C]
    const int*   __restrict__ label,   // [U,B]
    const int*   __restrict__ blankp,
    const float* __restrict__ emit,    // [B,T,LABT] (PRE only)
    float*       __restrict__ out)     // align [B,T,LABT] then loss [B], as float
{
    __shared__ int           s_ext[LABT];
    __shared__ int           s_idx[LABT];
    __shared__ unsigned char s_cond[LABT];
    __shared__ float         s_dp[2][LABT + 2];          // slots [0],[1] = permanent pad
    __shared__ unsigned char s_bp[(Tt / 4) * BPS];       // 2-bit deltas, 4 timesteps/byte
    __shared__ unsigned char s_path[Tt];
    __shared__ int           s_yl;
    __shared__ float         s_emit[4][LABT];            // async-staged emit ring (PRE)

    const int tid = threadIdx.x;
    const int b   = blockIdx.x;
    const int j   = tid;
    const int blank = *blankp;

    const float* lrow = logit + (size_t)b * Tt * (size_t)Cc;
    const float* erow = PRE ? (emit + (size_t)b * Tt * (size_t)LABT) : nullptr;

    // ---- extended labels, gather indices ----
    if (j < LABT) {
        int extv = blank;
        if (j & 1) {
            const int lv = label[(j >> 1) * Bb + b];
            extv = (lv == blank) ? -1 : lv;
        }
        s_ext[j] = extv;
        int idx = extv % Cc;
        if (idx < 0) idx += Cc;
        s_idx[j] = idx;
    }
    if (tid == 0) {
        s_dp[0][0] = PADV; s_dp[0][1] = PADV;
        s_dp[1][0] = PADV; s_dp[1][1] = PADV;
        int nv = 0;
        for (int u = 0; u < Uu; ++u) nv += (label[u * Bb + b] != blank) ? 1 : 0;
        s_yl = 2 * nv + 1;
    }
    __syncthreads();

    // ---- cond + dp0; prologue async copies of emit rows 1..3 (PRE) ----
    if (j < LABT) {
        const int sh2 = (j >= 2) ? s_ext[j - 2] : -1;    // shifted2 pads with -1
        s_cond[j] = (s_ext[j] == blank || s_ext[j] == sh2) ? 1 : 0;
        float d0 = PADV;
        if (j < 2) d0 = PRE ? erow[j] : lrow[s_idx[j]];  // emit[b,0,{0,1}]
        s_dp[0][2 + j] = d0;
        if (PRE) {
            async_copy_b32(&s_emit[1][j], erow + (size_t)1 * LABT + j);
            async_copy_b32(&s_emit[2][j], erow + (size_t)2 * LABT + j);
            async_copy_b32(&s_emit[3][j], erow + (size_t)3 * LABT + j);
        }
    }
    __syncthreads();

    const bool c = (j < LABT) ? (s_cond[j] != 0) : false;

    // ---- sequential DP over T ----
    int pp = 0;
    unsigned int pack = 0;                               // 2-bit deltas; t=0 slot stays 0 (identity bp0)
    float e_cur = 0.0f;
    if (!PRE && j < LABT) e_cur = lrow[(size_t)1 * Cc + s_idx[j]];

    for (int t = 1; t < Tt; ++t) {
        float e_next = 0.0f;
        if (!PRE) {
            if (j < LABT && (t + 1) < Tt)
                e_next = lrow[(size_t)(t + 1) * Cc + s_idx[j]];
            if (j < LABT && (t + 3) < Tt)
                __builtin_prefetch(lrow + (size_t)(t + 3) * Cc + s_idx[j], 0, 0);
        }

        if (PRE) {
            // Async loads complete in order: row t is done once at most the
            // two newer copies remain outstanding in this wave.
            const int rem = (Tt - 1) - t;
            if (rem >= 2)      asm volatile("s_wait_asynccnt 0x2" ::: "memory");
            else if (rem == 1) asm volatile("s_wait_asynccnt 0x1" ::: "memory");
            else               asm volatile("s_wait_asynccnt 0x0" ::: "memory");
        }
        __syncthreads();   // row t visible; dp[t-1] visible; iter t-1 reads retired

        if (PRE && j < LABT && (t + 3) < Tt)   // refill slot last read at iter t-1
            async_copy_b32(&s_emit[(t + 3) & 3][j], erow + (size_t)(t + 3) * LABT + j);

        if (j < LABT) {
            const float e   = PRE ? s_emit[t & 3][j] : e_cur;
            const float cur = s_dp[pp][2 + j];
            const float a1  = s_dp[pp][1 + j];           // dp_prev[j-1] (pad for j==0)
            const float a2  = s_dp[pp][0 + j];           // dp_prev[j-2] (pad for j<2)

            const float m2 = fmaxf(cur, a1);
            const float l2 = m2 + logf(expf(cur - m2) + expf(a1 - m2));
            const float m3 = fmaxf(m2, a2);
            const float l3 = m3 + logf(expf(cur - m3) + expf(a1 - m3) + expf(a2 - m3));

            // jnp.argmax first-max tie-breaking
            const int i1 = (cur >= a1) ? 0 : 1;
            const int i2 = (cur >= a1 && cur >= a2) ? 0 : ((a1 >= a2) ? 1 : 2);

            const float dpv = e + (c ? l2 : l3);
            const int   dlt = c ? i1 : i2;

            s_dp[pp ^ 1][2 + j] = dpv;
            pack |= ((unsigned)dlt) << (2 * (t & 3));
            if ((t & 3) == 3) {
                s_bp[(t >> 2) * BPS + j] = (unsigned char)pack;
                pack = 0;
            }
        }
        pp ^= 1;
        if (!PRE) e_cur = e_next;
    }
    __syncthreads();   // publish final dp buffer + last bp bytes

    // ---- loss + backtrack (LDS-resident) ----
    if (tid == 0) {
        const int yl = s_yl;
        int i1 = yl - 1; if (i1 < 0) i1 += LABT;         // JAX negative-index wrap
        int i2 = yl - 2; if (i2 < 0) i2 += LABT;
        const float d1 = s_dp[pp][2 + i1];
        const float d2 = s_dp[pp][2 + i2];
        const float mx = fmaxf(d1, d2);
        const float la = mx + log1pf(expf(fminf(d1, d2) - mx));   // logaddexp
        out[(size_t)Bb * Tt * LABT + b] = (-la) * 2.0f / (float)(yl - 1);

        int wi = (d1 > d2) ? i1 : i2;
        for (int t = Tt - 1; t >= 0; --t) {
            s_path[t] = (unsigned char)wi;
            const int dlt = (s_bp[(t >> 2) * BPS + wi] >> (2 * (t & 3))) & 3;
            wi -= dlt;
        }
    }
    __syncthreads();

    // ---- one-hot alignment writeout (fully rewrites d_out region each call) ----
    float* arow = out + (size_t)b * Tt * LABT;
    for (int idx = tid; idx < Tt * LABT; idx += blockDim.x) {
        const int t  = idx / LABT;
        const int jj = idx - t * LABT;
        arow[idx] = (jj == (int)s_path[t]) ? 1.0f : 0.0f;
    }
}

// ---------------------------------------------------------------------------
extern "C" void kernel_launch(void* const* d_in, const int* in_sizes, int n_in,
                              void* d_out, int out_size, void* d_ws, size_t ws_size,
                              hipStream_t stream)
{
    (void)in_sizes; (void)n_in; (void)out_size;
    const float* logit  = (const float*)d_in[0];
    const int*   label  = (const int*)d_in[1];
    const int*   blankp = (const int*)d_in[2];
    float*       out    = (float*)d_out;

    const size_t need = (size_t)Bb * Tt * LABT * sizeof(float);
    if (d_ws != nullptr && ws_size >= need) {
        float* emit = (float*)d_ws;
        const int total = Bb * Tt * LABT;
        gather_emit_kernel<<<(total + 255) / 256, 256, 0, stream>>>(logit, label, blankp, emit);
        ctc_fa_kernel<true><<<Bb, 256, 0, stream>>>(logit, label, blankp, emit, out);
    } else {
        ctc_fa_kernel<false><<<Bb, 256, 0, stream>>>(logit, label, blankp, nullptr, out);
    }
}